// tsail_sur_61297773248804
// MI455X (gfx1250) — compile-verified
//
#include <hip/hip_runtime.h>
#include <hip/hip_bf16.h>

typedef __attribute__((ext_vector_type(2))) float v2f;
typedef __attribute__((ext_vector_type(8))) float v8f;

#define N_NODES 100000
#define N_EDGES 1600000

// ---------------------------------------------------------------------------
// SpMM: agg[n,f] = sum_{e in row-range(n)} val[e] * x[col[e], f]
// Rows sorted -> per-node edge range via binary search (deterministic, no
// float atomics). One block per node; edge metadata staged once in LDS.
// Feature gathers are lane-contiguous -> coalesced, L2-resident.
// ---------------------------------------------------------------------------
__global__ void spmm_kernel(const float* __restrict__ x,
                            const int* __restrict__ erow,
                            const int* __restrict__ ecol,
                            const float* __restrict__ eval,
                            float* __restrict__ agg,
                            int N, int F, int E)
{
    const int node = blockIdx.x;
    __shared__ int s_lo, s_hi;
    __shared__ int   s_col[256];
    __shared__ float s_val[256];

    if (threadIdx.x == 0) {
        int lo = 0, hi = E;                       // lower_bound(erow, node)
        while (lo < hi) { int mid = (lo + hi) >> 1; if (erow[mid] < node) lo = mid + 1; else hi = mid; }
        s_lo = lo;
        int lo2 = lo, hi2 = E;                    // lower_bound(erow, node+1)
        while (lo2 < hi2) { int mid = (lo2 + hi2) >> 1; if (erow[mid] < node + 1) lo2 = mid + 1; else hi2 = mid; }
        s_hi = lo2;
    }
    __syncthreads();

    const int lo = s_lo, hi = s_hi;
    const int f = threadIdx.x;
    float acc = 0.0f;

    for (int base = lo; base < hi; base += 256) {
        int cnt = hi - base; if (cnt > 256) cnt = 256;
        if (threadIdx.x < cnt) {
            s_col[threadIdx.x] = ecol[base + threadIdx.x];
            s_val[threadIdx.x] = eval[base + threadIdx.x];
        }
        __syncthreads();
        if (f < F) {
            for (int e = 0; e < cnt; ++e)
                acc = fmaf(s_val[e], x[(size_t)s_col[e] * F + f], acc);
        }
        __syncthreads();
    }
    if (f < F) agg[(size_t)node * F + f] = acc;
}

// ---------------------------------------------------------------------------
// Dual fp32 WMMA GEMM: out = [relu]( A1@W1 (+ A2@W2) + b1 (+ b2) )
// A: N x K row-major, W: K x Fo row-major, out: N x Fo row-major.
// 256-thread block = 8 waves; each wave owns one 16x16 output tile, all 8
// share one LDS-staged, zero-padded 16-column weight panel (stride K+4 to
// dodge bank conflicts). N % 16 == 0 -> no row guards; column guard handled
// entirely by panel zero-padding, so the K-loop is branch-free:
//   global_load_b64 (A pair) + ds_load_b64 (B pair) + v_wmma_f32_16x16x4_f32
// K is a compile-time template constant (100/200/128) for scheduling freedom.
// ---------------------------------------------------------------------------
template <int K, bool HAS_A2>
__global__ void gemm_dual_wmma(const float* __restrict__ A1, const float* __restrict__ W1,
                               const float* __restrict__ A2, const float* __restrict__ W2,
                               const float* __restrict__ b1, const float* __restrict__ b2,
                               float* __restrict__ out,
                               int N, int Fo, int do_relu)
{
    constexpr int KS = K + 4;                 // LDS panel stride (floats)
    __shared__ float sW1[16 * KS];
    __shared__ float sW2[16 * KS];

    const int nt = blockIdx.y * 16;           // column-tile base

    // Stage zero-padded weight panels: sW[n_local][k] = W[k][nt+n_local] or 0
    for (int idx = threadIdx.x; idx < 16 * K; idx += 256) {
        const int nl = idx & 15;
        const int kk = idx >> 4;
        const int col = nt + nl;
        const bool ok = (col < Fo);
        sW1[nl * KS + kk] = ok ? W1[(size_t)kk * Fo + col] : 0.0f;
        if (HAS_A2)
            sW2[nl * KS + kk] = ok ? W2[(size_t)kk * Fo + col] : 0.0f;
    }
    __syncthreads();

    const int tid  = threadIdx.x;
    const int wave = tid >> 5;
    const int lane = tid & 31;
    const int half = lane >> 4;               // K sub-pair selector
    const int l15  = lane & 15;

    const int mt = (blockIdx.x * 8 + wave) * 16;
    if (mt >= N) return;                      // wave-uniform; N % 16 == 0

    const float* arow1 = A1 + (size_t)(mt + l15) * K + half * 2;
    const float* bp1   = sW1 + l15 * KS + half * 2;
    const float* arow2 = HAS_A2 ? (A2 + (size_t)(mt + l15) * K + half * 2) : nullptr;
    const float* bp2   = sW2 + l15 * KS + half * 2;

    v8f acc = {};
    for (int k0 = 0; k0 < K; k0 += 4) {
        v2f a = *(const v2f*)(arow1 + k0);
        v2f b = *(const v2f*)(bp1 + k0);
        acc = __builtin_amdgcn_wmma_f32_16x16x4_f32(false, a, false, b,
                                                    (short)0, acc, false, false);
    }
    if (HAS_A2) {
        for (int k0 = 0; k0 < K; k0 += 4) {
            v2f a = *(const v2f*)(arow2 + k0);
            v2f b = *(const v2f*)(bp2 + k0);
            acc = __builtin_amdgcn_wmma_f32_16x16x4_f32(false, a, false, b,
                                                        (short)0, acc, false, false);
        }
    }

    // Epilogue. C/D layout: acc[i] = D[mt + i + 8*half][nt + l15]
    const int ncol = nt + l15;
    if (ncol < Fo) {
        float bias = b1[ncol];
        if (HAS_A2) bias += b2[ncol];
        #pragma unroll
        for (int i = 0; i < 8; ++i) {
            const int r = mt + i + 8 * half;  // always < N
            float v = acc[i] + bias;
            if (do_relu) v = fmaxf(v, 0.0f);
            out[(size_t)r * Fo + ncol] = v;
        }
    }
}

// ---------------------------------------------------------------------------
// BatchNorm over axis 0 of t2 (N x 256): deterministic two-pass reduction.
// ---------------------------------------------------------------------------
__global__ void bn_partial(const float* __restrict__ t2,
                           float* __restrict__ ps, float* __restrict__ pq,
                           int N, int rowsPer)
{
    const int rb = blockIdx.x;          // 0..255
    const int c  = threadIdx.x;         // 0..255
    const int r0 = rb * rowsPer;
    int r1 = r0 + rowsPer; if (r1 > N) r1 = N;
    float s = 0.0f, q = 0.0f;
    for (int n = r0; n < r1; ++n) {
        float v = t2[(size_t)n * 256 + c];
        s += v; q += v * v;
    }
    ps[rb * 256 + c] = s;
    pq[rb * 256 + c] = q;
}

__global__ void bn_finalize(const float* __restrict__ ps, const float* __restrict__ pq,
                            const float* __restrict__ g, const float* __restrict__ bb,
                            float* __restrict__ scale, float* __restrict__ shift, int N)
{
    const int c = blockIdx.x;           // column
    const int t = threadIdx.x;          // row-block
    __shared__ float ss[256], sq[256];
    ss[t] = ps[t * 256 + c];
    sq[t] = pq[t * 256 + c];
    __syncthreads();
    for (int off = 128; off > 0; off >>= 1) {
        if (t < off) { ss[t] += ss[t + off]; sq[t] += sq[t + off]; }
        __syncthreads();
    }
    if (t == 0) {
        const float inv = 1.0f / (float)N;
        const float mu  = ss[0] * inv;
        const float var = sq[0] * inv - mu * mu;
        const float sc  = g[c] * rsqrtf(var + 1e-5f);
        scale[c] = sc;
        shift[c] = bb[c] - mu * sc;
    }
}

// ---------------------------------------------------------------------------
// Final head: out[n,:18] = relu(t2[n,:]*scale+shift) @ fc2b_W + fc2b_b
// 32 nodes per block; BN+ReLU'd rows and the 256x18 weight staged in LDS.
// ---------------------------------------------------------------------------
__global__ void bn_relu_fc2b(const float* __restrict__ t2,
                             const float* __restrict__ scale, const float* __restrict__ shift,
                             const float* __restrict__ Wb, const float* __restrict__ bb,
                             float* __restrict__ out, int N)
{
    __shared__ float sh[32 * 256];
    __shared__ float sW[256 * 18];
    const int base = blockIdx.x * 32;

    for (int i = threadIdx.x; i < 256 * 18; i += 256) sW[i] = Wb[i];
    for (int idx = threadIdx.x; idx < 32 * 256; idx += 256) {
        const int nl = idx >> 8;
        const int c  = idx & 255;
        const int n  = base + nl;
        float v = (n < N) ? t2[(size_t)n * 256 + c] : 0.0f;
        v = fmaf(v, scale[c], shift[c]);
        sh[idx] = fmaxf(v, 0.0f);
    }
    __syncthreads();

    for (int o = threadIdx.x; o < 32 * 18; o += 256) {
        const int nl = o / 18;
        const int j  = o - nl * 18;
        const int n  = base + nl;
        if (n < N) {
            float acc = bb[j];
            const float* row = &sh[nl * 256];
            #pragma unroll 8
            for (int c = 0; c < 256; ++c)
                acc = fmaf(row[c], sW[c * 18 + j], acc);
            out[(size_t)n * 18 + j] = acc;
        }
    }
}

// ---------------------------------------------------------------------------
extern "C" void kernel_launch(void* const* d_in, const int* in_sizes, int n_in,
                              void* d_out, int out_size, void* d_ws, size_t ws_size,
                              hipStream_t stream)
{
    const float* x     = (const float*)d_in[0];
    const int*   erow  = (const int*)  d_in[1];
    const int*   ecol  = (const int*)  d_in[2];
    const float* eval  = (const float*)d_in[3];
    const float* W0    = (const float*)d_in[4];
    const float* b0    = (const float*)d_in[5];
    const float* Ws0   = (const float*)d_in[6];
    const float* bs0   = (const float*)d_in[7];
    const float* W1    = (const float*)d_in[8];
    const float* b1    = (const float*)d_in[9];
    const float* Ws1   = (const float*)d_in[10];
    const float* bs1   = (const float*)d_in[11];
    const float* W2    = (const float*)d_in[12];
    const float* b2    = (const float*)d_in[13];
    const float* Ws2   = (const float*)d_in[14];
    const float* bs2   = (const float*)d_in[15];
    const float* fc1W  = (const float*)d_in[16];
    const float* fc1b  = (const float*)d_in[17];
    const float* fc2aW = (const float*)d_in[18];
    const float* fc2ab = (const float*)d_in[19];
    const float* bng   = (const float*)d_in[20];
    const float* bnb   = (const float*)d_in[21];
    const float* fc2bW = (const float*)d_in[22];
    const float* fc2bb = (const float*)d_in[23];

    const int N = N_NODES, E = N_EDGES;

    // Workspace layout (floats): A(N*256) | B(N*200) | C(N*128) | partials | scale/shift
    float* ws    = (float*)d_ws;
    float* bufA  = ws;                         // 25,600,000 fl
    float* bufB  = ws + 25600000;              // 20,000,000 fl
    float* bufC  = ws + 45600000;              // 12,800,000 fl
    float* psum  = ws + 58400000;              // 65,536 fl
    float* psq   = psum + 65536;               // 65,536 fl
    float* scal  = psq + 65536;                // 256 fl
    float* shft  = scal + 256;                 // 256 fl

    const int rowsPer  = (N + 255) / 256;      // 391
    const int rowBlks  = (N + 127) / 128;      // 8 row-tiles (waves) per block

    // GC layer 0: 100 -> 200
    spmm_kernel<<<N, 256, 0, stream>>>(x, erow, ecol, eval, bufA, N, 100, E);
    gemm_dual_wmma<100, true><<<dim3(rowBlks, (200 + 15) / 16), 256, 0, stream>>>(
        bufA, W0, x, Ws0, b0, bs0, bufB, N, 200, 1);

    // GC layer 1: 200 -> 128
    spmm_kernel<<<N, 256, 0, stream>>>(bufB, erow, ecol, eval, bufA, N, 200, E);
    gemm_dual_wmma<200, true><<<dim3(rowBlks, 128 / 16), 256, 0, stream>>>(
        bufA, W1, bufB, Ws1, b1, bs1, bufC, N, 128, 1);

    // GC layer 2: 128 -> 128
    spmm_kernel<<<N, 256, 0, stream>>>(bufC, erow, ecol, eval, bufA, N, 128, E);
    gemm_dual_wmma<128, true><<<dim3(rowBlks, 128 / 16), 256, 0, stream>>>(
        bufA, W2, bufC, Ws2, b2, bs2, bufB, N, 128, 1);

    // fc1 (128 -> 128, relu)
    gemm_dual_wmma<128, false><<<dim3(rowBlks, 128 / 16), 256, 0, stream>>>(
        bufB, fc1W, nullptr, nullptr, fc1b, nullptr, bufC, N, 128, 1);

    // fc2a (128 -> 256, no relu; BN follows)
    gemm_dual_wmma<128, false><<<dim3(rowBlks, 256 / 16), 256, 0, stream>>>(
        bufC, fc2aW, nullptr, nullptr, fc2ab, nullptr, bufA, N, 256, 0);

    // BatchNorm stats (deterministic two-pass)
    bn_partial<<<256, 256, 0, stream>>>(bufA, psum, psq, N, rowsPer);
    bn_finalize<<<256, 256, 0, stream>>>(psum, psq, bng, bnb, scal, shft, N);

    // BN + ReLU + fc2b (256 -> 18)
    bn_relu_fc2b<<<(N + 31) / 32, 256, 0, stream>>>(
        bufA, scal, shft, fc2bW, fc2bb, (float*)d_out, N);
}